// ChamferLoss_46832323395807
// MI455X (gfx1250) — compile-verified
//
#include <hip/hip_runtime.h>
#include <hip/hip_bf16.h>
#include <math.h>

typedef __attribute__((ext_vector_type(2))) float v2f;
typedef __attribute__((ext_vector_type(8))) float v8f;
typedef __attribute__((ext_vector_type(4))) int v4i;

#define KNN 8
#define TPB 256                    // 8 waves of 32
#define CHUNK 2048                 // points staged in LDS per pass (32 KB)
#define QPW 16                     // queries per wave (16 WMMA columns)
#define QUERIES_PER_WG ((TPB / 32) * QPW)   // 128

#if defined(__AMDGCN__) && __has_builtin(__builtin_amdgcn_global_load_async_to_lds_b128)
#define USE_ASYNC_LDS 1
typedef __attribute__((address_space(1))) v4i gv4i;  // global (device) pointee
typedef __attribute__((address_space(3))) v4i lv4i;  // LDS pointee
#endif

// ---------------------------------------------------------------------------
// Prep: pred = src + flow. Emit per cloud:
//   Q-format (x, y, z, |p|^2)        -- consumed as WMMA B columns / final sqrt
//   A-format (-2x, -2y, -2z, |p|^2)  -- consumed as WMMA A rows (no VALU fixup)
// ---------------------------------------------------------------------------
__global__ __launch_bounds__(TPB) void chamfer_prep_kernel(
    const float* __restrict__ src, const float* __restrict__ tgt,
    const float* __restrict__ flow,
    float4* __restrict__ predQ, float4* __restrict__ predA,
    float4* __restrict__ tgtQ, float4* __restrict__ tgtA, int total) {
  int i = blockIdx.x * blockDim.x + threadIdx.x;
  if (i >= total) return;
  float px = src[3 * i + 0] + flow[3 * i + 0];
  float py = src[3 * i + 1] + flow[3 * i + 1];
  float pz = src[3 * i + 2] + flow[3 * i + 2];
  float pn = px * px + py * py + pz * pz;
  predQ[i] = make_float4(px, py, pz, pn);
  predA[i] = make_float4(-2.0f * px, -2.0f * py, -2.0f * pz, pn);
  float tx = tgt[3 * i + 0];
  float ty = tgt[3 * i + 1];
  float tz = tgt[3 * i + 2];
  float tn = tx * tx + ty * ty + tz * tz;
  tgtQ[i] = make_float4(tx, ty, tz, tn);
  tgtA[i] = make_float4(-2.0f * tx, -2.0f * ty, -2.0f * tz, tn);
}

// ---------------------------------------------------------------------------
// KNN(K=8) Chamfer via V_WMMA_F32_16X16X4_F32.
//   D = |p|^2 - 2 q.p  per (point-row, query-col); each lane owns one query
//   column and gets 8 candidate distances per WMMA. Two tiles per iteration,
//   next tile's A fragments preloaded before the insertion network runs.
// ---------------------------------------------------------------------------
__global__ __launch_bounds__(TPB) void chamfer_knn_kernel(
    const float4* __restrict__ predQ, const float4* __restrict__ predA,
    const float4* __restrict__ tgtQ, const float4* __restrict__ tgtA,
    float* __restrict__ out, int n, float scale) {
  __shared__ v2f spts2[CHUNK * 2];  // point i -> [2i]=(K0,K1), [2i+1]=(K2,K3)

  const int b   = blockIdx.y;
  const int dir = blockIdx.z;
  const float4* __restrict__ Q  = (dir == 0 ? predQ : tgtQ) + (size_t)b * n;
  const float4* __restrict__ PA = (dir == 0 ? tgtA : predA) + (size_t)b * n;

  const int tid  = threadIdx.x;
  const int lane = tid & 31;
  const int wave = tid >> 5;
  const int col  = lane & 15;
  const bool hiHalf = lane >= 16;
  const int fragIdx = 2 * col + (hiHalf ? 1 : 0);  // v2f index within a tile

  const int qidx = blockIdx.x * QUERIES_PER_WG + wave * QPW + col;
  const float4 q = Q[qidx];

  v2f bvec;
  bvec.x = hiHalf ? q.z : q.x;   // K2 : K0
  bvec.y = hiHalf ? 1.0f : q.y;  // K3 : K1

  float heap[KNN];
#pragma unroll
  for (int j = 0; j < KNN; ++j) heap[j] = 3.0e38f;

  for (int c0 = 0; c0 < n; c0 += CHUNK) {
    __syncthreads();
#ifdef USE_ASYNC_LDS
#pragma unroll
    for (int k = 0; k < CHUNK / TPB; ++k) {
      gv4i* gsrc = (gv4i*)((v4i*)const_cast<float4*>(&PA[c0 + tid + k * TPB]));
      lv4i* ldst = (lv4i*)((v4i*)&spts2[2 * (tid + k * TPB)]);
      __builtin_amdgcn_global_load_async_to_lds_b128(gsrc, ldst, 0, 0);
    }
#if __has_builtin(__builtin_amdgcn_s_wait_asynccnt)
    __builtin_amdgcn_s_wait_asynccnt(0);
#else
    asm volatile("s_wait_asynccnt 0" ::: "memory");
#endif
#else
#pragma unroll
    for (int k = 0; k < CHUNK / TPB; ++k)
      ((float4*)spts2)[tid + k * TPB] = PA[c0 + tid + k * TPB];
#endif
    __syncthreads();
    if (c0 + CHUNK < n)
      __builtin_prefetch(&PA[c0 + CHUNK + tid], 0, 1);  // global_prefetch_b8

    v2f a0 = spts2[fragIdx];
    v2f a1 = spts2[fragIdx + 32];
    for (int it = 0; it < CHUNK / 32; ++it) {
      v8f cz0 = {};
      v8f d0 = __builtin_amdgcn_wmma_f32_16x16x4_f32(
          false, a0, false, bvec, (short)0, cz0, false, false);
      v8f cz1 = {};
      v8f d1 = __builtin_amdgcn_wmma_f32_16x16x4_f32(
          false, a1, false, bvec, (short)0, cz1, false, false);

      // Preload next pair of tiles while the insertion network runs.
      if (it + 1 < CHUNK / 32) {
        a0 = spts2[(it + 1) * 64 + fragIdx];
        a1 = spts2[(it + 1) * 64 + fragIdx + 32];
      }

#pragma unroll
      for (int r = 0; r < KNN; ++r) {
        float v = d0[r];
        if (__any(v < heap[KNN - 1])) {
          float x = v;
#pragma unroll
          for (int j = 0; j < KNN; ++j) {
            float lo = fminf(heap[j], x);
            float hv = fmaxf(heap[j], x);
            heap[j] = lo;
            x = hv;
          }
        }
      }
#pragma unroll
      for (int r = 0; r < KNN; ++r) {
        float v = d1[r];
        if (__any(v < heap[KNN - 1])) {
          float x = v;
#pragma unroll
          for (int j = 0; j < KNN; ++j) {
            float lo = fminf(heap[j], x);
            float hv = fmaxf(heap[j], x);
            heap[j] = lo;
            x = hv;
          }
        }
      }
    }
  }

  // Merge the two half-wave candidate lists for each query column.
  float other[KNN];
#pragma unroll
  for (int j = 0; j < KNN; ++j) other[j] = __shfl_xor(heap[j], 16, 32);
#pragma unroll
  for (int j = 0; j < KNN; ++j) {
    float x = other[j];
#pragma unroll
    for (int t = 0; t < KNN; ++t) {
      float lo = fminf(heap[t], x);
      float hv = fmaxf(heap[t], x);
      heap[t] = lo;
      x = hv;
    }
  }

  // dist per query = mean_k sqrt(s_k + |q|^2)
  float sum = 0.0f;
#pragma unroll
  for (int j = 0; j < KNN; ++j) sum += sqrtf(fmaxf(heap[j] + q.w, 0.0f));
  float contrib = hiHalf ? 0.0f : sum * (1.0f / KNN);

#pragma unroll
  for (int off = 16; off > 0; off >>= 1)
    contrib += __shfl_xor(contrib, off, 32);
  if (lane == 0) atomicAdd(out, contrib * scale);
}

// ---------------------------------------------------------------------------
extern "C" void kernel_launch(void* const* d_in, const int* in_sizes, int n_in,
                              void* d_out, int out_size, void* d_ws, size_t ws_size,
                              hipStream_t stream) {
  const float* src  = (const float*)d_in[0];  // pc_source [B,N,3]
  const float* tgt  = (const float*)d_in[1];  // pc_target [B,N,3]
  const float* flow = (const float*)d_in[2];  // pred_flow [B,N,3]

  const int total = in_sizes[0] / 3;  // B*N
  const int B = 4;
  const int n = total / B;            // 8192

  float4* predQ = (float4*)d_ws;
  float4* predA = predQ + total;
  float4* tgtQ  = predA + total;
  float4* tgtA  = tgtQ + total;

  (void)hipMemsetAsync(d_out, 0, sizeof(float), stream);

  chamfer_prep_kernel<<<(total + TPB - 1) / TPB, TPB, 0, stream>>>(
      src, tgt, flow, predQ, predA, tgtQ, tgtA, total);

  dim3 grid(n / QUERIES_PER_WG, B, 2);
  chamfer_knn_kernel<<<grid, TPB, 0, stream>>>(predQ, predA, tgtQ, tgtA,
                                               (float*)d_out, n,
                                               1.0f / (float)(B * n));
}